// Filter_1864015806874
// MI455X (gfx1250) — compile-verified
//
#include <hip/hip_runtime.h>

// ---------------------------------------------------------------------------
// Fused equivariant filter kernel for gfx1250 (MI455X).
//   out[b,n,d] = sum_k R[b,n,k,d] * dot(Y[b,n,k,:], tensor[b,k,d,:])
//   R = MLP(rbf) via two 32x32 fp32 GEMMs on V_WMMA_F32_16X16X4_F32.
// Memory-bound: rbf stream (67 MB) sets the floor (~3 us @ 23.3 TB/s);
// WMMA makes the 2.15 GFLOP of GEMM negligible and R is never materialized.
// ---------------------------------------------------------------------------

typedef float v2f __attribute__((ext_vector_type(2)));
typedef float v8f __attribute__((ext_vector_type(8)));

#define BB  2
#define NNP 512   // points == neighbors
#define CCH 32    // channels == RBF size

// D = A(16x4) * B(4x16) + C(16x16), all fp32, one wave.
__device__ __forceinline__ v8f wmma4(v2f a, v2f b, v8f c) {
  return __builtin_amdgcn_wmma_f32_16x16x4_f32(
      /*neg_a=*/false, a, /*neg_b=*/false, b,
      /*c_mod=*/(short)0, c, /*reuse_a=*/false, /*reuse_b=*/false);
}

__global__ __launch_bounds__(256) void filter_fused_kernel(
    const float* __restrict__ tensor,  // [B,N,C,3]
    const float* __restrict__ rbf,     // [B,N,N,32]
    const float* __restrict__ rij,     // [B,N,N,3]
    const float* __restrict__ W1,      // [32,32]  (m,l)
    const float* __restrict__ B1,      // [32]
    const float* __restrict__ W2,      // [32,32]  (d,m)
    const float* __restrict__ B2,      // [32]
    float* __restrict__ out)           // [B,N,C,1]
{
  // Per-wave H staging tile: 16 edges x 32 ch, row stride 34 (conflict-free,
  // keeps 8B alignment for ds_load_b64 of v2f fragments).
  __shared__ float h_lds[8 * 16 * 34];
  // Deterministic cross-wave reduction buffer: 16 partials x 32 channels.
  __shared__ float red[16 * 32];

  const int tid  = threadIdx.x;
  const int lane = tid & 31;
  const int wave = tid >> 5;
  const int col  = lane & 15;   // N-column of WMMA tiles / channel within tile
  const int half = lane >> 4;   // K split for A/B fragments per ISA layout

  const int b = blockIdx.x >> 9;          // B*N = 1024 blocks
  const int n = blockIdx.x & (NNP - 1);

  // -------------------------------------------------------------------
  // Weight fragments (B-matrix, K x N) held in registers for both layers.
  // B[k=l, n=m] = W[m, l]; lane layout mirrors the 16x4 fp32 A layout:
  //   VGPR0 (x): K = 4c + 2*half ; VGPR1 (y): K = 4c + 2*half + 1 ; N = col.
  // -------------------------------------------------------------------
  v2f w1f[2][8], w2f[2][8];
  float bias1[2], bias2[2];
#pragma unroll
  for (int t = 0; t < 2; ++t) {
    const int m = col + 16 * t;
    bias1[t] = B1[m];
    bias2[t] = B2[m];
#pragma unroll
    for (int c = 0; c < 8; ++c) {
      const int l0 = 4 * c + 2 * half;
      w1f[t][c].x = W1[m * 32 + l0];
      w1f[t][c].y = W1[m * 32 + l0 + 1];
      w2f[t][c].x = W2[m * 32 + l0];
      w2f[t][c].y = W2[m * 32 + l0 + 1];
    }
  }

  const float* rbf_rows = rbf + ((size_t)(b * NNP + n) * NNP) * 32;
  const float* rij_rows = rij + ((size_t)(b * NNP + n) * NNP) * 3;
  float* hw = &h_lds[wave * (16 * 34)];

  float p0 = 0.0f, p1 = 0.0f;   // per-lane partials for d=col and d=col+16

  // Each wave: 64 neighbors as 4 tiles of M=16 edges.
  for (int tt = 0; tt < 4; ++tt) {
    const int k_base = wave * 64 + tt * 16;

    // ---------------- Layer 1: H = relu(rbf * W1^T + b1) ----------------
    v8f acc0 = {};
    v8f acc1 = {};
    const float* arow = rbf_rows + (size_t)(k_base + col) * 32;  // lane's M row
#pragma unroll
    for (int c = 0; c < 8; ++c) {
      const int l0 = 4 * c + 2 * half;          // even -> 8B aligned
      v2f a = *(const v2f*)(arow + l0);         // global_load_b64
      acc0 = wmma4(a, w1f[0][c], acc0);
      acc1 = wmma4(a, w1f[1][c], acc1);
    }
    // C layout: VGPR r holds (edge = r + 8*half, channel = col + 16*t).
#pragma unroll
    for (int r = 0; r < 8; ++r) {
      const int e = r + 8 * half;
      hw[e * 34 + col]      = fmaxf(acc0[r] + bias1[0], 0.0f);
      hw[e * 34 + col + 16] = fmaxf(acc1[r] + bias1[1], 0.0f);
    }
    // Same-wave LDS ops are in-order; make the RAW explicit anyway.
    asm volatile("s_wait_dscnt 0" ::: "memory");

    // ---------------- Layer 2: R = H * W2^T + b2 ------------------------
    v8f r0 = {};
    v8f r1 = {};
#pragma unroll
    for (int c = 0; c < 8; ++c) {
      const int m0 = 4 * c + 2 * half;
      v2f a = *(const v2f*)(hw + col * 34 + m0);  // ds_load_b64, conflict-free
      r0 = wmma4(a, w2f[0][c], r0);
      r1 = wmma4(a, w2f[1][c], r1);
    }

    // ------------- Fused epilogue: p += (R + b2) * (Y . tensor) ---------
#pragma unroll
    for (int r = 0; r < 8; ++r) {
      const int k = k_base + r + 8 * half;
      const float* rp = rij_rows + (size_t)k * 3;   // broadcast across 16 lanes
      const float x = rp[0], y = rp[1], z = rp[2];
      const float r2 = x * x + y * y + z * z;
      // Y = rij * rsqrt(max(r2, EPS)); mask (sqrt(r2) < EPS) folded into scale.
      const float scl = (r2 >= 1e-16f) ? rsqrtf(fmaxf(r2, 1e-8f)) : 0.0f;
      const float* tp = tensor + ((size_t)(b * NNP + k) * CCH) * 3;
      const float* t0 = tp + col * 3;               // coalesced 192B per half
      const float* t1 = tp + (col + 16) * 3;
      const float s0 = (x * t0[0] + y * t0[1] + z * t0[2]) * scl;
      const float s1 = (x * t1[0] + y * t1[1] + z * t1[2]) * scl;
      p0 += (r0[r] + bias2[0]) * s0;
      p1 += (r1[r] + bias2[1]) * s1;
    }
  }

  // Deterministic reduction over the 16 (wave, half) partial rows.
  red[(wave * 2 + half) * 32 + col]      = p0;
  red[(wave * 2 + half) * 32 + col + 16] = p1;
  __syncthreads();
  if (tid < CCH) {
    float s = 0.0f;
#pragma unroll
    for (int i = 0; i < 16; ++i) s += red[i * 32 + tid];
    out[(size_t)(b * NNP + n) * CCH + tid] = s;  // full overwrite (RO == 1)
  }
}

extern "C" void kernel_launch(void* const* d_in, const int* in_sizes, int n_in,
                              void* d_out, int out_size, void* d_ws, size_t ws_size,
                              hipStream_t stream) {
  (void)in_sizes; (void)n_in; (void)d_ws; (void)ws_size; (void)out_size;
  const float* tensor = (const float*)d_in[0];
  const float* rbf    = (const float*)d_in[1];
  const float* rij    = (const float*)d_in[2];
  const float* W1     = (const float*)d_in[3];
  const float* B1     = (const float*)d_in[4];
  const float* W2     = (const float*)d_in[5];
  const float* B2     = (const float*)d_in[6];
  float* out = (float*)d_out;

  dim3 grid(BB * NNP);   // one workgroup per (b, n) row
  dim3 block(256);       // 8 wave32
  filter_fused_kernel<<<grid, block, 0, stream>>>(tensor, rbf, rij, W1, B1, W2,
                                                  B2, out);
}